// MaxChamferDistance_80212809220557
// MI455X (gfx1250) — compile-verified
//
#include <hip/hip_runtime.h>
#include <hip/hip_bf16.h>

typedef __attribute__((ext_vector_type(2))) float v2f;
typedef __attribute__((ext_vector_type(8))) float v8f;

#define NPTS             4096
#define CHUNK            2048      // columns staged per LDS pass (32 KB)
#define WGSIZE           256
#define WAVES_PER_WG     8
#define ROWS_PER_WAVE    16
#define ROWS_PER_WG      (WAVES_PER_WG * ROWS_PER_WAVE)   // 128
#define BLOCKS_PER_BATCH (NPTS / ROWS_PER_WG)             // 32
#define NBATCH           16

// Directed Chamfer term via WMMA with K=4 trick:
//   A[m] = (-2px, -2py, -2pz, 1),  B[n] = (qx, qy, qz, ||q||^2)
//   => A x B = -2<p,q> + ||q||^2 ;  C = ||p||^2 (loop-invariant accumulator)
//   => D[m][n] = ||p_m - q_n||^2 in a single v_wmma_f32_16x16x4_f32.
__global__ __launch_bounds__(WGSIZE)
void chamfer_directed_wmma(const float* __restrict__ P,
                           const float* __restrict__ Q,
                           float* __restrict__ partials) {
    __shared__ float qcol[CHUNK * 4];   // (qx, qy, qz, q2) per column, 32 KB
    __shared__ float wavesums[WAVES_PER_WG];

    const int b      = blockIdx.y;
    const int rowblk = blockIdx.x;
    const int tid    = threadIdx.x;
    const int wave   = tid >> 5;
    const int lane   = tid & 31;
    const int l16    = lane & 15;
    const int hi     = lane >> 4;          // 0: lanes 0-15, 1: lanes 16-31

    const float* Qb = Q + (size_t)b * NPTS * 3;
    const float* Pb = P + (size_t)b * NPTS * 3;

    // ---- A tile: 16 rows of P. A layout: lane l -> (M=l%16, K=2*hi+vgpr) ----
    const int   row = rowblk * ROWS_PER_WG + wave * ROWS_PER_WAVE + l16;
    const float px  = Pb[row * 3 + 0];
    const float py  = Pb[row * 3 + 1];
    const float pz  = Pb[row * 3 + 2];
    const float x2  = px * px + py * py + pz * pz;

    v2f a;
    a.x = (hi == 0) ? (-2.0f * px) : (-2.0f * pz);   // K=0 / K=2
    a.y = (hi == 0) ? (-2.0f * py) : 1.0f;           // K=1 / K=3 (y2 pickup)

    // Loop-invariant C: C[v] = x2 of row (v + 8*hi). Lane (v+8*hi) holds it.
    v8f cm;
    #pragma unroll
    for (int v = 0; v < 8; ++v)
        cm[v] = __shfl(x2, v + 8 * hi, 32);

    float rowmin[8];
    #pragma unroll
    for (int v = 0; v < 8; ++v) rowmin[v] = 3.402823466e38f;

    // ---- Sweep all columns in two LDS chunks --------------------------------
    for (int c0 = 0; c0 < NPTS; c0 += CHUNK) {
        __syncthreads();   // all waves done reading previous chunk
        #pragma unroll
        for (int p = tid; p < CHUNK; p += WGSIZE) {
            const float* qs = Qb + (size_t)(c0 + p) * 3;
            const float qxv = qs[0], qyv = qs[1], qzv = qs[2];
            qcol[p * 4 + 0] = qxv;
            qcol[p * 4 + 1] = qyv;
            qcol[p * 4 + 2] = qzv;
            qcol[p * 4 + 3] = qxv * qxv + qyv * qyv + qzv * qzv;
        }
        __syncthreads();

        for (int j0 = 0; j0 < CHUNK; j0 += 16) {
            const int c = j0 + l16;
            // B layout: lane l -> (K=2*hi+vgpr, N=l%16):
            //   hi=0 reads (qx,qy), hi=1 reads (qz,q2) -> one ds_load_b64.
            const v2f bm = *reinterpret_cast<const v2f*>(&qcol[c * 4 + hi * 2]);

            const v8f d = __builtin_amdgcn_wmma_f32_16x16x4_f32(
                /*neg_a=*/false, a, /*neg_b=*/false, bm,
                /*c_mod=*/(short)0, cm, /*reuse_a=*/false, /*reuse_b=*/false);

            #pragma unroll
            for (int v = 0; v < 8; ++v) rowmin[v] = fminf(rowmin[v], d[v]);
        }
    }

    // ---- Row mins: reduce over the 16 N-lanes of each half-wave -------------
    #pragma unroll
    for (int off = 8; off > 0; off >>= 1) {
        #pragma unroll
        for (int v = 0; v < 8; ++v)
            rowmin[v] = fminf(rowmin[v], __shfl_xor(rowmin[v], off, 16));
    }

    // Each half-wave holds identical mins for its 8 rows; sum both halves.
    float s = 0.0f;
    #pragma unroll
    for (int v = 0; v < 8; ++v) s += rowmin[v];
    s += __shfl_xor(s, 16, 32);

    if (lane == 0) wavesums[wave] = s;
    __syncthreads();

    if (tid == 0) {
        float t = 0.0f;
        #pragma unroll
        for (int w = 0; w < WAVES_PER_WG; ++w) t += wavesums[w];
        partials[b * BLOCKS_PER_BATCH + rowblk] = t;
    }
}

// Deterministic final reduction: mean over rows, per-batch max of the two
// directed terms, mean over batch.
__global__ void chamfer_finalize(const float* __restrict__ part0,
                                 const float* __restrict__ part1,
                                 float* __restrict__ out) {
    if (threadIdx.x == 0) {
        float total = 0.0f;
        for (int b = 0; b < NBATCH; ++b) {
            float s0 = 0.0f, s1 = 0.0f;
            for (int k = 0; k < BLOCKS_PER_BATCH; ++k) {
                s0 += part0[b * BLOCKS_PER_BATCH + k];
                s1 += part1[b * BLOCKS_PER_BATCH + k];
            }
            const float d0 = s0 * (1.0f / (float)NPTS);  // x -> y
            const float d1 = s1 * (1.0f / (float)NPTS);  // y -> x
            total += fmaxf(d0, d1);
        }
        out[0] = total * (1.0f / (float)NBATCH);
    }
}

extern "C" void kernel_launch(void* const* d_in, const int* in_sizes, int n_in,
                              void* d_out, int out_size, void* d_ws, size_t ws_size,
                              hipStream_t stream) {
    (void)in_sizes; (void)n_in; (void)out_size; (void)ws_size;
    const float* x = (const float*)d_in[0];   // [16, 4096, 3]
    const float* y = (const float*)d_in[1];   // [16, 4096, 3]
    float* out = (float*)d_out;

    float* part0 = (float*)d_ws;                       // [16][32]
    float* part1 = part0 + NBATCH * BLOCKS_PER_BATCH;  // [16][32]

    dim3 grid(BLOCKS_PER_BATCH, NBATCH);
    chamfer_directed_wmma<<<grid, WGSIZE, 0, stream>>>(x, y, part0);
    chamfer_directed_wmma<<<grid, WGSIZE, 0, stream>>>(y, x, part1);
    chamfer_finalize<<<1, 32, 0, stream>>>(part0, part1, out);
}